// YvParallelScan_72112500900642
// MI455X (gfx1250) — compile-verified
//
#include <hip/hip_runtime.h>
#include <cstdint>

// Mamba-style chunked diagonal SSM scan, specialized for:
//   D_MODEL=1024, D_STATE=64, CHUNK=256, BATCH=4, SEQ_LEN=4096
// Layout: one lane <-> one channel d. B_t[n], C_t[n] are wave-uniform
// (broadcast LDS reads); h[n,d] and A[n,d] live in VGPRs as float2 pairs,
// and the inner loop is pinned to v_pk_fma_f32 / v_pk_mul_f32 (VOP3P packed
// f32, 2x VALU rate). u is streamed through LDS with gfx1250 async-to-LDS
// copies, double buffered in 16-timestep slabs.

#define D_MODEL  1024
#define D_STATE  64
#define CHUNK    256
#define BATCH    4
#define SEQ_LEN  4096
#define N_CHUNKS (SEQ_LEN / CHUNK)   // 16
#define D_TILE   256                  // channels per workgroup (== blockDim)
#define N_DTILES (D_MODEL / D_TILE)   // 4
#define T_SLAB   16                   // timesteps per u-slab in LDS
#define N_SLABS  (CHUNK / T_SLAB)     // 16
#define NPAIR    (D_STATE / 2)        // 32 float2 state pairs per thread

typedef float vf2 __attribute__((ext_vector_type(2)));
typedef float vf4 __attribute__((ext_vector_type(4)));
typedef int   v4i __attribute__((ext_vector_type(4)));

#if defined(__AMDGCN__) && __has_builtin(__builtin_amdgcn_global_load_async_to_lds_b128) && \
    __has_builtin(__builtin_amdgcn_s_wait_asynccnt)
#define HAVE_ASYNC 1
#else
#define HAVE_ASYNC 0
#endif

#if defined(__AMDGCN__) && __has_builtin(__builtin_amdgcn_sched_barrier)
#define SCHED_FENCE() __builtin_amdgcn_sched_barrier(0)
#else
#define SCHED_FENCE()
#endif

#if HAVE_ASYNC
// Builtin signature (from hipcc diagnostic): param0 is AS1 (global) v4i*,
// param1 is the LDS-side v4i*; 4 args total: (src, dst, imm offset, imm cpol).
typedef __attribute__((address_space(1))) v4i g_v4i;
typedef __attribute__((address_space(3))) v4i l_v4i;
// 16-byte async copy global->LDS (ASYNCcnt-tracked, no VGPR data movement).
__device__ __forceinline__ void async_copy16(const void* g, void* l) {
  __builtin_amdgcn_global_load_async_to_lds_b128(
      (g_v4i*)(unsigned long long)(uintptr_t)g,
      (l_v4i*)(unsigned)(uintptr_t)l,
      0 /*offset*/, 0 /*cpol*/);
}
#endif

// ---- Packed f32 math pinned to VOP3P (2 lanes of FMA per instruction) ----
#if defined(__AMDGCN__)
__device__ __forceinline__ vf2 pk_fma(vf2 a, vf2 b, vf2 c) {
  vf2 d;
  asm("v_pk_fma_f32 %0, %1, %2, %3" : "=v"(d) : "v"(a), "v"(b), "v"(c));
  return d;
}
__device__ __forceinline__ vf2 pk_mul(vf2 a, vf2 b) {
  vf2 d;
  asm("v_pk_mul_f32 %0, %1, %2" : "=v"(d) : "v"(a), "v"(b));
  return d;
}
#else
__device__ __forceinline__ vf2 pk_fma(vf2 a, vf2 b, vf2 c) {
  return __builtin_elementwise_fma(a, b, c);
}
__device__ __forceinline__ vf2 pk_mul(vf2 a, vf2 b) { return a * b; }
#endif

__global__ __launch_bounds__(D_TILE)
void YvParallelScan_72112500900642_kernel(const float* __restrict__ u,
                                          const float* __restrict__ Bm,
                                          const float* __restrict__ Cm,
                                          const float* __restrict__ Alog,
                                          float* __restrict__ out) {
  // Dynamic LDS: sB[256][64] (64KB) | sC[256][64] (64KB) | sU[2][16][256] (32KB)
  extern __shared__ float smem[];
  float* sB = smem;
  float* sC = smem + CHUNK * D_STATE;
  float* sU = smem + 2 * CHUNK * D_STATE;

  const int tid   = threadIdx.x;
  const int wg    = blockIdx.x;            // 0..255
  const int b     = wg >> 6;               // batch
  const int chunk = (wg >> 2) & (N_CHUNKS - 1);
  const int dt    = wg & (N_DTILES - 1);   // d-tile
  const int d     = dt * D_TILE + tid;     // this lane's channel
  const long t0   = (long)b * SEQ_LEN + (long)chunk * CHUNK;

  const float* uBase = u   + t0 * D_MODEL + dt * D_TILE;
  const float* bBase = Bm  + t0 * D_STATE;   // chunk's B: 256x64, contiguous
  const float* cBase = Cm  + t0 * D_STATE;   // chunk's C: 256x64, contiguous
  float*       yBase = out + t0 * D_MODEL + d;

  // ---- Stage the whole chunk's B and C into LDS (async, one-time) ----
  #pragma unroll
  for (int i = 0; i < (CHUNK * D_STATE) / (4 * D_TILE); ++i) {  // 16 iters
    const int f4 = (i * D_TILE + tid) * 4;  // float offset, 16B-aligned
#if HAVE_ASYNC
    async_copy16(bBase + f4, sB + f4);
    async_copy16(cBase + f4, sC + f4);
#else
    *(vf4*)(sB + f4) = *(const vf4*)(bBase + f4);
    *(vf4*)(sC + f4) = *(const vf4*)(cBase + f4);
#endif
  }

  // ---- Per-thread A[n] = -exp(A_log[n, d]) and h[n] = 0, packed in pairs ----
  vf2 A2[NPAIR], h2[NPAIR];
  #pragma unroll
  for (int n = 0; n < NPAIR; ++n) {
    const float a0 = Alog[(size_t)(2 * n + 0) * D_MODEL + d];
    const float a1 = Alog[(size_t)(2 * n + 1) * D_MODEL + d];
    A2[n].x = -expf(a0);
    A2[n].y = -expf(a1);
    h2[n].x = 0.0f;
    h2[n].y = 0.0f;
  }

  // ---- u slab loader: T_SLAB rows x D_TILE floats, 4 x b128 per thread ----
  const int col  = (tid & 63) * 4;  // float column within the 256-wide row
  const int row0 = (tid >> 6) * 4;  // 4 consecutive rows per thread group
  auto load_slab = [&](int slab, int buf) {
    const float* src = uBase + (size_t)slab * T_SLAB * D_MODEL;
    float*       dst = sU + buf * (T_SLAB * D_TILE);
    #pragma unroll
    for (int r = 0; r < 4; ++r) {
      const int row = row0 + r;
#if HAVE_ASYNC
      async_copy16(src + (size_t)row * D_MODEL + col, dst + row * D_TILE + col);
#else
      *(vf4*)(dst + row * D_TILE + col) = *(const vf4*)(src + (size_t)row * D_MODEL + col);
#endif
    }
  };

  load_slab(0, 0);
#if HAVE_ASYNC
  __builtin_amdgcn_s_wait_asynccnt(0);
#endif
  __syncthreads();

  // ---- Main scan: 96 packed VOP3P ops per timestep per lane ----
  for (int slab = 0; slab < N_SLABS; ++slab) {
    const int buf = slab & 1;
    if (slab + 1 < N_SLABS) load_slab(slab + 1, buf ^ 1);  // async prefetch

    const float* uS = sU + buf * (T_SLAB * D_TILE);
    #pragma unroll 2
    for (int tt = 0; tt < T_SLAB; ++tt) {
      const int   t  = slab * T_SLAB + tt;
      const float x  = uS[tt * D_TILE + tid];   // ds_load_b32, conflict-free
      vf2 x2;
      x2.x = x;
      x2.y = x;
      const vf2* bRow = (const vf2*)(sB + t * D_STATE);  // wave-uniform rows
      const vf2* cRow = (const vf2*)(sC + t * D_STATE);  // (LDS broadcast)
      vf2 yA = {0.0f, 0.0f};
      vf2 yB = {0.0f, 0.0f};
      #pragma unroll
      for (int g = 0; g < 4; ++g) {            // 4 groups of 8 pairs
        vf2 bb[8], cc[8];                       // B/C register working set
        #pragma unroll
        for (int p = 0; p < 8; ++p) {
          bb[p] = bRow[g * 8 + p];
          cc[p] = cRow[g * 8 + p];
        }
        #pragma unroll
        for (int p = 0; p < 8; ++p) {
          const int n  = g * 8 + p;
          const vf2 bx = pk_mul(bb[p], x2);        // v_pk_mul_f32
          h2[n] = pk_fma(A2[n], h2[n], bx);        // v_pk_fma_f32
          if (p & 1) yB = pk_fma(cc[p], h2[n], yB);// v_pk_fma_f32
          else       yA = pk_fma(cc[p], h2[n], yA);
        }
        SCHED_FENCE();  // keep LDS loads grouped; cap register pressure
      }
      const float y = (yA.x + yA.y) + (yB.x + yB.y);
      __builtin_nontemporal_store(y, yBase + (size_t)t * D_MODEL);
    }

#if HAVE_ASYNC
    __builtin_amdgcn_s_wait_asynccnt(0);  // own async loads for next slab done
#endif
    __syncthreads();                      // everyone's loads visible
  }
}

extern "C" void kernel_launch(void* const* d_in, const int* in_sizes, int n_in,
                              void* d_out, int out_size, void* d_ws, size_t ws_size,
                              hipStream_t stream) {
  (void)in_sizes; (void)n_in; (void)out_size; (void)d_ws; (void)ws_size;
  const float* u    = (const float*)d_in[0];
  // d_in[1] = delta: unused by the reference forward.
  const float* Bm   = (const float*)d_in[2];
  const float* Cm   = (const float*)d_in[3];
  const float* Alog = (const float*)d_in[4];
  float* out = (float*)d_out;

  const dim3 grid(BATCH * N_CHUNKS * N_DTILES);  // 256 workgroups
  const dim3 block(D_TILE);                      // 256 threads = 8 wave32
  const size_t smemBytes =
      (size_t)(2 * CHUNK * D_STATE + 2 * T_SLAB * D_TILE) * sizeof(float);  // 160 KB

  YvParallelScan_72112500900642_kernel<<<grid, block, smemBytes, stream>>>(
      u, Bm, Cm, Alog, out);
}